// GCN_Net2_1529008358069
// MI455X (gfx1250) — compile-verified
//
#include <hip/hip_runtime.h>
#include <hip/hip_bf16.h>
#include <math.h>

#define NNODES   100000
#define NEDGES   1600000
#define NGRAPHS  512
#define EPSBN    1e-5f

typedef __attribute__((ext_vector_type(2))) float v2f;
typedef __attribute__((ext_vector_type(8))) float v8f;

// ------------------------------------------------------------------
// f32 WMMA GEMM with compile-time strides:
//   C[M,N] = act(A[M,K] @ Bt^T + bias),  Bt is B transposed: Bt[N][K].
// block = 128 threads (4 waves); each wave computes one 16x16 tile.
// grid.x = M/16, grid.y = N/64. Full K unroll -> immediate-offset loads.
// A fragment (16x4 f32): lane L holds M=L%16, K=2*(L/16)+{0,1}  -> float2
// B fragment (4x16 f32): lane L holds N=L%16, K=2*(L/16)+{0,1}  -> float2 (from Bt)
// C fragment: VGPR v, lane L -> M = v + 8*(L/16), N = L%16
// ------------------------------------------------------------------
template<int LDA, int LDC, int K, int RELU>
__global__ void k_gemm_wmma(const float* __restrict__ A,
                            const float* __restrict__ Bt,
                            float* __restrict__ C,
                            const float* __restrict__ bias) {
  const int lane = threadIdx.x & 31;
  const int wave = threadIdx.x >> 5;
  const int m    = lane & 15;
  const int kh   = (lane >> 4) << 1;        // 0 or 2
  const int r0   = blockIdx.x * 16;
  const int n0   = blockIdx.y * 64 + wave * 16;

  const float2* Af = reinterpret_cast<const float2*>(A  + (size_t)(r0 + m) * LDA + kh);
  const float2* Bf = reinterpret_cast<const float2*>(Bt + (size_t)(n0 + m) * K   + kh);
  v8f acc = {};
#pragma unroll
  for (int k = 0; k < K; k += 4) {
    float2 af = Af[k >> 1];                 // A[r0+m][k+kh .. +1]
    float2 bf = Bf[k >> 1];                 // B[k+kh .. +1][n0+m]
    v2f a; a.x = af.x; a.y = af.y;
    v2f b; b.x = bf.x; b.y = bf.y;
    acc = __builtin_amdgcn_wmma_f32_16x16x4_f32(false, a, false, b,
                                                (short)0, acc, false, false);
  }
  const int mbase = r0 + ((lane >> 4) << 3);
  const int ncol  = n0 + m;
  const float bv  = bias ? bias[ncol] : 0.0f;
#pragma unroll
  for (int v = 0; v < 8; ++v) {
    float val = acc[v] + bv;
    if (RELU) val = fmaxf(val, 0.0f);
    C[(size_t)(mbase + v) * LDC + ncol] = val;
  }
}

// Wt[n*KD + k] = W[k*ND + n]   (W: [KD,ND])
__global__ void k_transpose(const float* __restrict__ W, float* __restrict__ Wt,
                            int KD, int ND) {
  int t = blockIdx.x * blockDim.x + threadIdx.x;
  if (t >= KD * ND) return;
  int k = t / ND, n = t % ND;
  Wt[n * KD + k] = W[t];
}

// Layer-0 GEMM: Cin = 1 -> outer product x[i] * W0[0][j]
__global__ void k_gemm0(const float* __restrict__ x, const float* __restrict__ W0,
                        float* __restrict__ out) {
  int idx = blockIdx.x * blockDim.x + threadIdx.x;   // N*64 threads exactly
  out[idx] = x[idx >> 6] * W0[idx & 63];
}

// In-degree over edge targets (self-loops added later)
__global__ void k_degree(const int* __restrict__ col, float* __restrict__ counts) {
  int e = blockIdx.x * blockDim.x + threadIdx.x;
  if (e < NEDGES) atomicAdd(&counts[col[e]], 1.0f);
}

// counts += 1 (self-loop); dinv = counts^-0.5
__global__ void k_countsfix(float* __restrict__ counts, float* __restrict__ dinv) {
  int i = blockIdx.x * blockDim.x + threadIdx.x;
  if (i < NNODES) {
    float c = counts[i] + 1.0f;
    counts[i] = c;
    dinv[i] = rsqrtf(c);
  }
}

// One wave per edge: gather h_lin[row][0..63] (as 32 x float2), scale,
// scatter-add into agg[col][0..63]. agg stays L2-resident (25.6 MB << 192 MB).
__global__ void k_scatter(const int* __restrict__ row, const int* __restrict__ col,
                          const float* __restrict__ dinv,
                          const float* __restrict__ hlin, float* __restrict__ agg) {
  const int lane = threadIdx.x & 31;
  const int e = blockIdx.x * 8 + (threadIdx.x >> 5);
  if (e >= NEDGES) return;
  const int r = row[e], c = col[e];
  const float nrm = dinv[r] * dinv[c];
  const float2 h = reinterpret_cast<const float2*>(hlin + (size_t)r * 64)[lane];
  float* dst = agg + (size_t)c * 64 + 2 * lane;
  atomicAdd(dst,     nrm * h.x);
  atomicAdd(dst + 1, nrm * h.y);
}

// agg = (agg + dinv^2 * hlin)/counts + bias  (self-loop folded in, in place),
// and accumulate per-channel sum / sumsq via LDS then global atomics.
__global__ void k_finalize(float* __restrict__ agg, const float* __restrict__ hlin,
                           const float* __restrict__ counts, const float* __restrict__ dinv,
                           const float* __restrict__ bias, float* __restrict__ stats) {
  __shared__ float ssum[64];
  __shared__ float ssq[64];
  if (threadIdx.x < 64) { ssum[threadIdx.x] = 0.0f; ssq[threadIdx.x] = 0.0f; }
  __syncthreads();
  const int idx = blockIdx.x * blockDim.x + threadIdx.x;   // N*64 threads exactly
  const int i = idx >> 6, j = idx & 63;
  const float d = dinv[i];
  const float v = (agg[idx] + d * d * hlin[idx]) / counts[i] + bias[j];
  agg[idx] = v;
  atomicAdd(&ssum[j], v);
  atomicAdd(&ssq[j], v * v);
  __syncthreads();
  if (threadIdx.x < 64) {
    atomicAdd(&stats[threadIdx.x],      ssum[threadIdx.x]);
    atomicAdd(&stats[64 + threadIdx.x], ssq[threadIdx.x]);
  }
}

// Fold batchnorm into per-channel scale/shift (train mode: biased var)
__global__ void k_bnstats(const float* __restrict__ stats, const float* __restrict__ gamma,
                          const float* __restrict__ beta, float* __restrict__ ss) {
  int j = threadIdx.x;   // 64 threads
  const float invn = 1.0f / (float)NNODES;
  float mu  = stats[j] * invn;
  float var = stats[64 + j] * invn - mu * mu;
  float inv = rsqrtf(var + EPSBN);
  ss[j]      = gamma[j] * inv;
  ss[64 + j] = beta[j] - gamma[j] * inv * mu;
}

// h = relu(h*scale + shift), in place
__global__ void k_bnrelu(float* __restrict__ h, const float* __restrict__ ss) {
  int idx = blockIdx.x * blockDim.x + threadIdx.x;   // N*64 exactly
  int j = idx & 63;
  h[idx] = fmaxf(h[idx] * ss[j] + ss[64 + j], 0.0f);
}

// Global mean pool (sums + counts via atomics)
__global__ void k_pool(const float* __restrict__ h, const int* __restrict__ batch,
                       float* __restrict__ pooled, float* __restrict__ pcnt) {
  int idx = blockIdx.x * blockDim.x + threadIdx.x;   // N*64 exactly
  int i = idx >> 6, j = idx & 63;
  int g = batch[i];
  atomicAdd(&pooled[(size_t)g * 64 + j], h[idx]);
  if (j == 0) atomicAdd(&pcnt[g], 1.0f);
}

__global__ void k_poolavg(float* __restrict__ pooled, const float* __restrict__ pcnt) {
  int idx = blockIdx.x * blockDim.x + threadIdx.x;   // 512*64
  pooled[idx] /= fmaxf(pcnt[idx >> 6], 1.0f);
}

// fc2: [512,128] @ [128,10] + b  (N=10 not tile-shaped -> plain loop)
__global__ void k_fc2(const float* __restrict__ a, const float* __restrict__ w,
                      const float* __restrict__ b, float* __restrict__ out) {
  int t = blockIdx.x * blockDim.x + threadIdx.x;
  if (t >= NGRAPHS * 10) return;
  int g = t / 10, o = t % 10;
  float s = b[o];
  const float* ar = a + (size_t)g * 128;
  for (int k = 0; k < 128; ++k) s += ar[k] * w[k * 10 + o];
  out[t] = s;
}

extern "C" void kernel_launch(void* const* d_in, const int* in_sizes, int n_in,
                              void* d_out, int out_size, void* d_ws, size_t ws_size,
                              hipStream_t stream) {
  (void)in_sizes; (void)n_in; (void)out_size; (void)ws_size;
  const float* x     = (const float*)d_in[0];
  const int*   eidx  = (const int*)d_in[1];
  const int*   batch = (const int*)d_in[2];
  const float* Ws[4] = {(const float*)d_in[3],  (const float*)d_in[4],
                        (const float*)d_in[5],  (const float*)d_in[6]};
  const float* bs[4] = {(const float*)d_in[7],  (const float*)d_in[8],
                        (const float*)d_in[9],  (const float*)d_in[10]};
  const float* gs[4] = {(const float*)d_in[11], (const float*)d_in[12],
                        (const float*)d_in[13], (const float*)d_in[14]};
  const float* be[4] = {(const float*)d_in[15], (const float*)d_in[16],
                        (const float*)d_in[17], (const float*)d_in[18]};
  const float* fc1w = (const float*)d_in[19];
  const float* fc1b = (const float*)d_in[20];
  const float* fc2w = (const float*)d_in[21];
  const float* fc2b = (const float*)d_in[22];

  const int* row = eidx;
  const int* col = eidx + NEDGES;

  // Workspace carve-up (two ping-pong node buffers + small scratch)
  char* p = (char*)d_ws;
  const size_t nodeBytes = (size_t)NNODES * 64 * sizeof(float);
  float* b0     = (float*)p; p += nodeBytes;                 // h / agg (result buf)
  float* b1     = (float*)p; p += nodeBytes;                 // h_lin scratch
  float* counts = (float*)p; p += (size_t)NNODES * sizeof(float);
  float* dinv   = (float*)p; p += (size_t)NNODES * sizeof(float);
  float* stats  = (float*)p; p += 128 * sizeof(float);
  float* ss     = (float*)p; p += 128 * sizeof(float);
  float* pooled = (float*)p; p += (size_t)NGRAPHS * 64 * sizeof(float);
  float* pcnt   = (float*)p; p += (size_t)NGRAPHS * sizeof(float);
  float* fc1o   = (float*)p; p += (size_t)NGRAPHS * 128 * sizeof(float);
  float* wt     = (float*)p; p += (size_t)128 * 64 * sizeof(float);  // transposed W

  // Degrees (edges + self-loop) and dinv
  hipMemsetAsync(counts, 0, (size_t)NNODES * sizeof(float), stream);
  k_degree<<<(NEDGES + 255) / 256, 256, 0, stream>>>(col, counts);
  k_countsfix<<<(NNODES + 255) / 256, 256, 0, stream>>>(counts, dinv);

  const int elemBlocks256 = NNODES * 64 / 256;   // 25000
  const int elemBlocks512 = NNODES * 64 / 512;   // 12500
  for (int l = 0; l < 4; ++l) {
    if (l == 0) {
      k_gemm0<<<elemBlocks256, 256, 0, stream>>>(x, Ws[0], b1);
    } else {
      k_transpose<<<(64 * 64 + 255) / 256, 256, 0, stream>>>(Ws[l], wt, 64, 64);
      k_gemm_wmma<64, 64, 64, 0><<<dim3(NNODES / 16, 1), 128, 0, stream>>>(
          b0, wt, b1, nullptr);
    }
    hipMemsetAsync(b0, 0, nodeBytes, stream);
    k_scatter<<<NEDGES / 8, 256, 0, stream>>>(row, col, dinv, b1, b0);
    hipMemsetAsync(stats, 0, 128 * sizeof(float), stream);
    k_finalize<<<elemBlocks512, 512, 0, stream>>>(b0, b1, counts, dinv, bs[l], stats);
    k_bnstats<<<1, 64, 0, stream>>>(stats, gs[l], be[l], ss);
    k_bnrelu<<<elemBlocks256, 256, 0, stream>>>(b0, ss);
  }

  // Pool + MLP head
  hipMemsetAsync(pooled, 0, ((size_t)NGRAPHS * 64 + NGRAPHS) * sizeof(float), stream);
  k_pool<<<elemBlocks256, 256, 0, stream>>>(b0, batch, pooled, pcnt);
  k_poolavg<<<NGRAPHS * 64 / 256, 256, 0, stream>>>(pooled, pcnt);
  k_transpose<<<(64 * 128 + 255) / 256, 256, 0, stream>>>(fc1w, wt, 64, 128);
  k_gemm_wmma<64, 128, 64, 1><<<dim3(NGRAPHS / 16, 2), 128, 0, stream>>>(
      pooled, wt, fc1o, fc1b);
  k_fc2<<<(NGRAPHS * 10 + 255) / 256, 256, 0, stream>>>(fc1o, fc2w, fc2b, (float*)d_out);
}